// GraphFeatureEncoder_1116691497082
// MI455X (gfx1250) — compile-verified
//
#include <hip/hip_runtime.h>
#include <hip/hip_bf16.h>
#include <math.h>

typedef __attribute__((ext_vector_type(2))) float v2f;
typedef __attribute__((ext_vector_type(8))) float v8f;

#define OUTW 256   // H * C
#define CCH  64    // out channels
#define NH   4     // heads

// ---------------------------------------------------------------------------
// Dense GEMM  XW[n,256] = X[n,K] @ W[K,256] using V_WMMA_F32_16X16X4_F32.
// One wave32 owns one 16x16 output tile; K consumed 4 at a time.
// A lane layout (16x4 f32): lanes 0-15 -> K=kb+0,kb+1 ; lanes 16-31 -> K=kb+2,kb+3, M=lane%16
// B lane layout (4x16 f32): mirrored with N=lane%16.
// C/D: VGPR r -> rows (r, r+8) split across lane halves.
// ---------------------------------------------------------------------------
__global__ void gemm_wmma_f32(const float* __restrict__ X,
                              const float* __restrict__ W,
                              float* __restrict__ XW,
                              int n, int K) {
  const int lane = threadIdx.x & 31;
  const int wave = threadIdx.x >> 5;
  const int tilesN = OUTW / 16;                 // 16
  const int tilesM = (n + 15) >> 4;
  int tile = blockIdx.x * (blockDim.x >> 5) + wave;
  if (tile >= tilesM * tilesN) return;
  const int tm = (tile / tilesN) * 16;
  const int tn = (tile % tilesN) * 16;
  const int half = lane >> 4;                   // 0 or 1
  const int l16  = lane & 15;
  int m = tm + l16; if (m >= n) m = n - 1;      // clamp loads; stores guarded
  const int nn = tn + l16;

  v8f acc = {};
  for (int kb = 0; kb < K; kb += 4) {
    const int k0 = kb + half * 2;
    v2f a, b;
    a.x = X[m * K + k0];
    a.y = X[m * K + k0 + 1];
    b.x = W[k0 * OUTW + nn];
    b.y = W[(k0 + 1) * OUTW + nn];
    acc = __builtin_amdgcn_wmma_f32_16x16x4_f32(false, a, false, b,
                                                (short)0, acc, false, false);
  }
#pragma unroll
  for (int r = 0; r < 8; ++r) {
    const int row = tm + r + half * 8;
    if (row < n) XW[row * OUTW + tn + l16] = acc[r];
  }
}

// ---------------------------------------------------------------------------
// Per-edge attention: logits[h] = sum_k (x[dst,k]-x[src,k]) * U[k,h] + c[h],
// softmax over 4 heads. Edges [0,E) real, [E, E+N) self loops.
// ---------------------------------------------------------------------------
__global__ void att_kernel(const float* __restrict__ X,
                           const int* __restrict__ srcE,
                           const int* __restrict__ dstE,
                           const float* __restrict__ U,
                           const float* __restrict__ cvec,
                           float* __restrict__ att,
                           int K, int Ereal, int Etot) {
  int e = blockIdx.x * blockDim.x + threadIdx.x;
  if (e >= Etot) return;
  int s, d;
  if (e < Ereal) { s = srcE[e]; d = dstE[e]; }
  else           { s = d = e - Ereal; }
  float lg[NH];
#pragma unroll
  for (int h = 0; h < NH; ++h) lg[h] = cvec[h];
  for (int k = 0; k < K; ++k) {
    const float diff = X[d * K + k] - X[s * K + k];
#pragma unroll
    for (int h = 0; h < NH; ++h) lg[h] = fmaf(diff, U[k * NH + h], lg[h]);
  }
  const float mx = fmaxf(fmaxf(lg[0], lg[1]), fmaxf(lg[2], lg[3]));
  float sum = 0.f;
#pragma unroll
  for (int h = 0; h < NH; ++h) { lg[h] = __expf(lg[h] - mx); sum += lg[h]; }
  const float inv = 1.f / sum;
#pragma unroll
  for (int h = 0; h < NH; ++h) att[e * NH + h] = lg[h] * inv;
}

// ---------------------------------------------------------------------------
// Scatter: msg[e,c] = sum_h att[e,h] * xw[src[e], h*64 + c]; atomicAdd to agg[dst].
// 64 threads per edge (one per channel).
// ---------------------------------------------------------------------------
__global__ void scatter_kernel(const float* __restrict__ XW,
                               const float* __restrict__ att,
                               const int* __restrict__ srcE,
                               const int* __restrict__ dstE,
                               float* __restrict__ agg,
                               int Ereal, int Etot) {
  const int gid = blockIdx.x * blockDim.x + threadIdx.x;
  const int e = gid >> 6;
  const int c = gid & 63;
  if (e >= Etot) return;
  int s, d;
  if (e < Ereal) { s = srcE[e]; d = dstE[e]; }
  else           { s = d = e - Ereal; }
  const float a0 = att[e * NH + 0], a1 = att[e * NH + 1];
  const float a2 = att[e * NH + 2], a3 = att[e * NH + 3];
  const float* xw = XW + (size_t)s * OUTW;
  float v = a0 * xw[c] + a1 * xw[CCH + c] + a2 * xw[2 * CCH + c] + a3 * xw[3 * CCH + c];
  atomicAdd(&agg[(size_t)d * CCH + c], v);
}

__global__ void count_kernel(const int* __restrict__ dstE,
                             float* __restrict__ cnt, int Ereal, int Etot) {
  int e = blockIdx.x * blockDim.x + threadIdx.x;
  if (e >= Etot) return;
  const int d = (e < Ereal) ? dstE[e] : e - Ereal;
  atomicAdd(&cnt[d], 1.0f);
}

// ---------------------------------------------------------------------------
// out = relu(agg/cnt + bias); accumulate per-channel sum & sumsq via LDS atomics
// then one global atomic per channel per block (stats[0:64]=sum, [64:128]=sumsq).
// ---------------------------------------------------------------------------
__global__ void finalize_relu_stats(const float* __restrict__ agg,
                                    const float* __restrict__ cnt,
                                    const float* __restrict__ bias,
                                    float* __restrict__ out,
                                    float* __restrict__ stats, int n) {
  __shared__ float ssum[CCH];
  __shared__ float ssq[CCH];
  const int t = threadIdx.x;
  if (t < CCH) { ssum[t] = 0.f; ssq[t] = 0.f; }
  __syncthreads();
  const int gid = blockIdx.x * blockDim.x + t;
  const int i = gid >> 6;
  const int c = gid & 63;
  if (i < n) {
    float v = agg[(size_t)i * CCH + c] / cnt[i] + bias[c];
    v = fmaxf(v, 0.f);
    out[(size_t)i * CCH + c] = v;
    atomicAdd(&ssum[c], v);
    atomicAdd(&ssq[c], v * v);
  }
  __syncthreads();
  if (t < CCH) {
    atomicAdd(&stats[t], ssum[t]);
    atomicAdd(&stats[CCH + t], ssq[t]);
  }
}

__global__ void bn_apply(float* __restrict__ h,
                         const float* __restrict__ stats,
                         const float* __restrict__ g,
                         const float* __restrict__ be, int n) {
  const int gid = blockIdx.x * blockDim.x + threadIdx.x;
  const int i = gid >> 6;
  const int c = gid & 63;
  if (i >= n) return;
  const float invN = 1.0f / (float)n;
  const float mean = stats[c] * invN;
  const float var  = stats[CCH + c] * invN - mean * mean;
  const float v = h[(size_t)i * CCH + c];
  h[(size_t)i * CCH + c] = (v - mean) * rsqrtf(var + 1e-5f) * g[c] + be[c];
}

// Final layer: out = agg/cnt + bias (no relu/BN).
__global__ void finalize_out(const float* __restrict__ agg,
                             const float* __restrict__ cnt,
                             const float* __restrict__ bias,
                             float* __restrict__ out, int n) {
  const int gid = blockIdx.x * blockDim.x + threadIdx.x;
  const int i = gid >> 6;
  const int c = gid & 63;
  if (i >= n) return;
  out[(size_t)i * CCH + c] = agg[(size_t)i * CCH + c] / cnt[i] + bias[c];
}

static inline int cdiv(int a, int b) { return (a + b - 1) / b; }

extern "C" void kernel_launch(void* const* d_in, const int* in_sizes, int n_in,
                              void* d_out, int out_size, void* d_ws, size_t ws_size,
                              hipStream_t stream) {
  (void)n_in; (void)out_size; (void)ws_size;

  const float* x  = (const float*)d_in[0];
  const int*   ei = (const int*)d_in[1];
  const float* w0 = (const float*)d_in[2];
  const float* u0 = (const float*)d_in[3];
  const float* c0 = (const float*)d_in[4];
  const float* b0 = (const float*)d_in[5];
  const float* w1 = (const float*)d_in[6];
  const float* u1 = (const float*)d_in[7];
  const float* c1 = (const float*)d_in[8];
  const float* b1 = (const float*)d_in[9];
  const float* w2 = (const float*)d_in[10];
  const float* u2 = (const float*)d_in[11];
  const float* c2 = (const float*)d_in[12];
  const float* b2 = (const float*)d_in[13];
  const float* g0 = (const float*)d_in[14];
  const float* be0 = (const float*)d_in[15];
  const float* g1 = (const float*)d_in[16];
  const float* be1 = (const float*)d_in[17];

  const int K0 = in_sizes[2] / OUTW;      // 32
  const int K1 = in_sizes[6] / OUTW;      // 64
  const int N  = in_sizes[0] / K0;        // 25000
  const int E  = in_sizes[1] / 2;         // 200000
  const int Etot = E + N;                 // edges + self loops
  const int* srcE = ei;
  const int* dstE = ei + E;

  // --- workspace carve-up (floats) ---
  float* ws   = (float*)d_ws;
  float* xw   = ws;                 size_t off = (size_t)N * OUTW;
  float* agg  = ws + off;           off += (size_t)N * CCH;
  float* cnt  = ws + off;           off += (size_t)N;
  float* att  = ws + off;           off += (size_t)Etot * NH;
  float* hA   = ws + off;           off += (size_t)N * CCH;
  float* hB   = ws + off;           off += (size_t)N * CCH;
  float* stats = ws + off;          // 128 floats

  const int THREADS = 256;
  const int tilesM = cdiv(N, 16);
  const int gemmBlocks = cdiv(tilesM * (OUTW / 16), THREADS / 32);
  const int edgeBlocks = cdiv(Etot, THREADS);
  const int edgeChBlocks = cdiv(Etot * CCH, THREADS);
  const int nodeChBlocks = cdiv(N * CCH, THREADS);

  // in-degree counts (layer invariant)
  hipMemsetAsync(cnt, 0, (size_t)N * sizeof(float), stream);
  count_kernel<<<edgeBlocks, THREADS, 0, stream>>>(dstE, cnt, E, Etot);

  // ---------------- layer 0 (K=32): x -> hA ----------------
  gemm_wmma_f32<<<gemmBlocks, THREADS, 0, stream>>>(x, w0, xw, N, K0);
  att_kernel<<<edgeBlocks, THREADS, 0, stream>>>(x, srcE, dstE, u0, c0, att, K0, E, Etot);
  hipMemsetAsync(agg, 0, (size_t)N * CCH * sizeof(float), stream);
  hipMemsetAsync(stats, 0, 2 * CCH * sizeof(float), stream);
  scatter_kernel<<<edgeChBlocks, THREADS, 0, stream>>>(xw, att, srcE, dstE, agg, E, Etot);
  finalize_relu_stats<<<nodeChBlocks, THREADS, 0, stream>>>(agg, cnt, b0, hA, stats, N);
  bn_apply<<<nodeChBlocks, THREADS, 0, stream>>>(hA, stats, g0, be0, N);

  // ---------------- layer 1 (K=64): hA -> hB ----------------
  gemm_wmma_f32<<<gemmBlocks, THREADS, 0, stream>>>(hA, w1, xw, N, K1);
  att_kernel<<<edgeBlocks, THREADS, 0, stream>>>(hA, srcE, dstE, u1, c1, att, K1, E, Etot);
  hipMemsetAsync(agg, 0, (size_t)N * CCH * sizeof(float), stream);
  hipMemsetAsync(stats, 0, 2 * CCH * sizeof(float), stream);
  scatter_kernel<<<edgeChBlocks, THREADS, 0, stream>>>(xw, att, srcE, dstE, agg, E, Etot);
  finalize_relu_stats<<<nodeChBlocks, THREADS, 0, stream>>>(agg, cnt, b1, hB, stats, N);
  bn_apply<<<nodeChBlocks, THREADS, 0, stream>>>(hB, stats, g1, be1, N);

  // ---------------- layer 2 (K=64): hB -> d_out ----------------
  gemm_wmma_f32<<<gemmBlocks, THREADS, 0, stream>>>(hB, w2, xw, N, K1);
  att_kernel<<<edgeBlocks, THREADS, 0, stream>>>(hB, srcE, dstE, u2, c2, att, K1, E, Etot);
  hipMemsetAsync(agg, 0, (size_t)N * CCH * sizeof(float), stream);
  scatter_kernel<<<edgeChBlocks, THREADS, 0, stream>>>(xw, att, srcE, dstE, agg, E, Etot);
  finalize_out<<<nodeChBlocks, THREADS, 0, stream>>>(agg, cnt, b2, (float*)d_out, N);
}